// MANetwork_PLT_17987323036108
// MI455X (gfx1250) — compile-verified
//
#include <hip/hip_runtime.h>
#include <hip/hip_bf16.h>
#include <stdint.h>

// Problem constants (from reference)
#define B_   64
#define S_   256
#define H_   768
#define A_   8
#define GNUM_ 8192
#define GSZ_  64
#define E_   256
#define TOPK_ 10
#define C_   (TOPK_ * GSZ_)   // 640
#define M_   (B_ * A_)        // 512

typedef float v2f __attribute__((ext_vector_type(2)));
typedef float v8f __attribute__((ext_vector_type(8)));

// ---------------------------------------------------------------------------
// Async global->LDS staging (gfx1250 GLOBAL_LOAD_ASYNC_TO_LDS_B128, ASYNCcnt)
// with compile-safe fallback to synchronous float4 staging.
// Builtin signature (from hipcc diagnostic): param1 = int4 addrspace(1)*,
// param2 = int4 addrspace(3)*, then imm offset, imm cpol.
// ---------------------------------------------------------------------------
#if defined(__AMDGCN__) && __has_builtin(__builtin_amdgcn_global_load_async_to_lds_b128)
#define HAVE_ASYNC_LDS 1
typedef int v4i __attribute__((vector_size(16)));
typedef __attribute__((address_space(1))) v4i gv4i_t;
typedef __attribute__((address_space(3))) v4i lv4i_t;
#else
#define HAVE_ASYNC_LDS 0
#endif

__device__ __forceinline__ void copy16_to_lds(const float* __restrict__ g, float* l) {
#if HAVE_ASYNC_LDS
  // global VA bits are identical in generic and addrspace(1); LDS needs a real
  // addrspacecast (aperture-based generic representation).
  __builtin_amdgcn_global_load_async_to_lds_b128(
      (gv4i_t*)(uintptr_t)g, (lv4i_t*)l, 0, 0);
#else
  *(float4*)l = *(const float4*)g;
#endif
}

__device__ __forceinline__ void wait_lds_copies() {
#if HAVE_ASYNC_LDS
#if __has_builtin(__builtin_amdgcn_s_wait_asynccnt)
  __builtin_amdgcn_s_wait_asynccnt(0);
#else
  asm volatile("s_wait_asynccnt 0" ::: "memory");
#endif
#endif
}

// ---------------------------------------------------------------------------
// Kernel 1: fused multi-head attention pooling
//   atten[b,a,s] = <inputs[b,s,:], W_att[a,:]> + b_att[a]
//   p = softmax over s;   out[b,a,h] = sum_s p[a,s] * inputs[b,s,h]
// one block per batch element, 256 threads (8 wave32)
// ---------------------------------------------------------------------------
__global__ __launch_bounds__(256) void attention_kernel(
    const float* __restrict__ inputs,  // [B,S,H]
    const float* __restrict__ W_att,   // [A,H]
    const float* __restrict__ b_att,   // [A]
    float* __restrict__ out)           // [B*A, H]
{
  __shared__ float att[A_ * S_];   // 8 KB
  const int b = blockIdx.x;
  const int t = threadIdx.x;

  // ---- phase 1: logits, thread t handles s = t for all 8 heads
  {
    const float* xrow = inputs + ((size_t)b * S_ + t) * H_;
    float acc[A_];
#pragma unroll
    for (int a = 0; a < A_; ++a) acc[a] = 0.f;
    for (int h = 0; h < H_; h += 4) {
      float4 x = *(const float4*)(xrow + h);
#pragma unroll
      for (int a = 0; a < A_; ++a) {
        const float* w = W_att + a * H_ + h;
        acc[a] += x.x * w[0] + x.y * w[1] + x.z * w[2] + x.w * w[3];
      }
    }
#pragma unroll
    for (int a = 0; a < A_; ++a) att[a * S_ + t] = acc[a] + b_att[a];
  }
  __syncthreads();

  // ---- phase 2: softmax over s; wave w owns head a = w (wave32)
  {
    const int wave = t >> 5, lane = t & 31;
    const int a = wave;
    float mx = -3.402823466e38f;
    for (int s = lane; s < S_; s += 32) mx = fmaxf(mx, att[a * S_ + s]);
#pragma unroll
    for (int off = 16; off >= 1; off >>= 1) mx = fmaxf(mx, __shfl_xor(mx, off, 32));
    float sum = 0.f;
    for (int s = lane; s < S_; s += 32) {
      float e = __expf(att[a * S_ + s] - mx);
      att[a * S_ + s] = e;
      sum += e;
    }
#pragma unroll
    for (int off = 16; off >= 1; off >>= 1) sum += __shfl_xor(sum, off, 32);
    const float inv = 1.0f / sum;
    for (int s = lane; s < S_; s += 32) att[a * S_ + s] *= inv;
  }
  __syncthreads();

  // ---- phase 3: weighted sum; thread t owns h = t, t+256, t+512 for all heads
  {
    float acc[3][A_];
#pragma unroll
    for (int j = 0; j < 3; ++j)
#pragma unroll
      for (int a = 0; a < A_; ++a) acc[j][a] = 0.f;

    for (int s = 0; s < S_; ++s) {
      const float* xrow = inputs + ((size_t)b * S_ + s) * H_;
      const float x0 = xrow[t];
      const float x1 = xrow[t + 256];
      const float x2 = xrow[t + 512];
#pragma unroll
      for (int a = 0; a < A_; ++a) {
        const float p = att[a * S_ + s];
        acc[0][a] += p * x0;
        acc[1][a] += p * x1;
        acc[2][a] += p * x2;
      }
    }
#pragma unroll
    for (int a = 0; a < A_; ++a)
#pragma unroll
      for (int j = 0; j < 3; ++j)
        out[((size_t)(b * A_ + a)) * H_ + j * 256 + t] = acc[j][a];
  }
}

// ---------------------------------------------------------------------------
// WMMA fp32 GEMM:  C[m,n] = sum_k A[m,k] * Bmat[n,k]  (+ bias[n])
//   MODE 0: store full C [M x N]
//   MODE 1: store max over consecutive groups of 8 rows (heads) -> [(M/8) x N]
// Block = 128 threads (4 waves); each wave owns one 16x16 output tile,
// block covers 16 (M) x 64 (N). K chunked by 128 through double-buffered LDS
// filled with async global->LDS copies; compute V_WMMA_F32_16X16X4_F32.
// ---------------------------------------------------------------------------
template <int MODE>
__global__ __launch_bounds__(128) void gemm_wmma_f32(
    const float* __restrict__ Amat,  // [M x K] row-major
    const float* __restrict__ Bmat,  // [N x K] row-major
    const float* __restrict__ bias,  // [N]
    float* __restrict__ Cout,
    int Mdim, int Ndim, int Kdim)
{
  constexpr int KT = 128;
  constexpr int LP = KT + 4;  // pitch 132 floats (528B): 16B aligned, bank-safe
  __shared__ __align__(16) float As[2][16 * LP];
  __shared__ __align__(16) float Bs[2][64 * LP];

  const int t = threadIdx.x;
  const int wave = t >> 5;
  const int lane = t & 31;
  const int m0 = blockIdx.y * 16;
  const int g0 = blockIdx.x * 64;
  const int nloc = wave * 16;
  const int lm = lane & 15;          // m (for A) / n (for B,D) within tile
  const int khalf = (lane >> 4) * 2; // K sub-offset per half-wave

  v8f acc = {};

  auto issue_tiles = [&](int buf, int kc) {
    // A tile: 16 rows x 32 vec4
    for (int i = t; i < 16 * (KT / 4); i += 128) {
      const int r = i >> 5, kv = i & 31;
      copy16_to_lds(&Amat[(size_t)(m0 + r) * Kdim + kc + kv * 4],
                    &As[buf][r * LP + kv * 4]);
    }
    // B tile: 64 rows x 32 vec4
    for (int i = t; i < 64 * (KT / 4); i += 128) {
      const int r = i >> 5, kv = i & 31;
      copy16_to_lds(&Bmat[(size_t)(g0 + r) * Kdim + kc + kv * 4],
                    &Bs[buf][r * LP + kv * 4]);
    }
  };

  const int nch = Kdim / KT;
  issue_tiles(0, 0);

  for (int c = 0; c < nch; ++c) {
    wait_lds_copies();
    __syncthreads();                       // chunk c resident for all waves
    if (c + 1 < nch) issue_tiles((c + 1) & 1, (c + 1) * KT);

    const float* arow = &As[c & 1][lm * LP];
    const float* brow = &Bs[c & 1][(nloc + lm) * LP];
#pragma unroll
    for (int k = 0; k < KT; k += 4) {
      v2f a, b;
      a[0] = arow[k + khalf];
      a[1] = arow[k + khalf + 1];
      b[0] = brow[k + khalf];
      b[1] = brow[k + khalf + 1];
      acc = __builtin_amdgcn_wmma_f32_16x16x4_f32(false, a, false, b,
                                                  (short)0, acc, false, false);
    }
    __syncthreads();                       // done reading buffer c&1
  }

  const int n = g0 + nloc + lm;
  const float bv = bias ? bias[n] : 0.f;

  if (MODE == 0) {
    // D layout: lane holds column n; VGPR v holds row m = v + 8*(lane>=16)
#pragma unroll
    for (int v = 0; v < 8; ++v) {
      const int m = m0 + v + ((lane >> 4) << 3);
      Cout[(size_t)m * Ndim + n] = acc[v] + bv;
    }
  } else {
    // rows m0..m0+7 = (batch m0/8, heads 0..7) live in lanes 0-15,
    // rows m0+8..m0+15 = next batch in lanes 16-31 -> free max over heads
    float mx = acc[0];
#pragma unroll
    for (int v = 1; v < 8; ++v) mx = fmaxf(mx, acc[v]);
    const int brow = (m0 >> 3) + (lane >> 4);
    Cout[(size_t)brow * Ndim + n] = mx + bv;
  }
}

// ---------------------------------------------------------------------------
// Kernel: top-10 groups per batch row + candidate expansion + sigmoid scores
// one block per batch element; matches jax.lax.top_k order (desc, low idx tie)
// ---------------------------------------------------------------------------
__global__ __launch_bounds__(256) void topk_kernel(
    const float* __restrict__ c_out,   // [B, GNUM]
    const int* __restrict__ group_y,   // [GNUM, GSZ]
    int* __restrict__ cand_out,        // [B, C]
    float* __restrict__ scores_out)    // [B, C]
{
  __shared__ float sval[256];
  __shared__ int sidx[256];
  __shared__ int chosen[TOPK_];
  __shared__ float chval[TOPK_];
  const int b = blockIdx.x;
  const int t = threadIdx.x;
  const float* row = c_out + (size_t)b * GNUM_;

  for (int it = 0; it < TOPK_; ++it) {
    float best = -3.402823466e38f;
    int bidx = 0x7fffffff;
    for (int g = t; g < GNUM_; g += 256) {
      bool skip = false;
      for (int j = 0; j < it; ++j)
        if (chosen[j] == g) skip = true;
      const float v = row[g];
      if (!skip && (v > best || (v == best && g < bidx))) { best = v; bidx = g; }
    }
    sval[t] = best;
    sidx[t] = bidx;
    __syncthreads();
    for (int stride = 128; stride >= 1; stride >>= 1) {
      if (t < stride) {
        const float v2 = sval[t + stride];
        const int i2 = sidx[t + stride];
        if (v2 > sval[t] || (v2 == sval[t] && i2 < sidx[t])) { sval[t] = v2; sidx[t] = i2; }
      }
      __syncthreads();
    }
    if (t == 0) { chosen[it] = sidx[0]; chval[it] = sval[0]; }
    __syncthreads();
  }

  for (int i = t; i < C_; i += 256) {
    const int tk = i >> 6;      // which of the top-10
    const int j = i & 63;       // member within group
    const int grp = chosen[tk];
    cand_out[(size_t)b * C_ + i] = group_y[grp * GSZ_ + j];
    scores_out[(size_t)b * C_ + i] = 1.0f / (1.0f + __expf(-chval[tk]));
  }
}

// ---------------------------------------------------------------------------
// Kernel: label scoring via WMMA with gathered embeddings
//   logits[b,a,c] = <embed_table[cand[b,c],:], emb[b*8+a,:]>  (E=256)
//   l_out[b,c] = max over a
// A-tile = emb[b] (8 rows padded to 16 with zeros); B-tile = 64 gathered rows
// staged with async global->LDS copies (double buffered over E chunks).
// Block = 128 threads (4 waves / 4 c-tiles); grid = (C/64, B)
// ---------------------------------------------------------------------------
__global__ __launch_bounds__(128) void label_score_kernel(
    const float* __restrict__ emb,          // [M_ x E_]
    const float* __restrict__ embed_table,  // [LABELS x E_]
    const int* __restrict__ candidates,     // [B x C]
    float* __restrict__ l_out)              // [B x C]
{
  constexpr int KT = 128;
  constexpr int LP = KT + 4;
  __shared__ __align__(16) float Em[2][16 * LP];
  __shared__ __align__(16) float Cs[2][64 * LP];

  const int t = threadIdx.x;
  const int wave = t >> 5;
  const int lane = t & 31;
  const int b = blockIdx.y;
  const int c0 = blockIdx.x * 64;
  const int lm = lane & 15;
  const int khalf = (lane >> 4) * 2;

  v8f acc = {};

  // zero the padding rows (8..15) of both A buffers once
  for (int i = t; i < 8 * LP * 2; i += 128) {
    const int buf = i >= 8 * LP;
    const int j = buf ? i - 8 * LP : i;
    Em[buf][(8 + j / LP) * LP + (j % LP)] = 0.f;
  }

  auto issue_tiles = [&](int buf, int kc) {
    // emb rows 0..7 (8 rows x 32 vec4 = 256 vec4)
    for (int i = t; i < 8 * (KT / 4); i += 128) {
      const int r = i >> 5, kv = i & 31;
      copy16_to_lds(&emb[(size_t)(b * A_ + r) * E_ + kc + kv * 4],
                    &Em[buf][r * LP + kv * 4]);
    }
    // gathered embedding rows: 64 rows x 32 vec4
    for (int i = t; i < 64 * (KT / 4); i += 128) {
      const int r = i >> 5, kv = i & 31;
      const int label = candidates[b * C_ + c0 + r];
      copy16_to_lds(&embed_table[(size_t)label * E_ + kc + kv * 4],
                    &Cs[buf][r * LP + kv * 4]);
    }
  };

  const int nch = E_ / KT;  // 2
  issue_tiles(0, 0);

  for (int c = 0; c < nch; ++c) {
    wait_lds_copies();
    __syncthreads();
    if (c + 1 < nch) issue_tiles((c + 1) & 1, (c + 1) * KT);

    const float* arow = &Em[c & 1][lm * LP];
    const float* brow = &Cs[c & 1][(wave * 16 + lm) * LP];
#pragma unroll
    for (int k = 0; k < KT; k += 4) {
      v2f a, bb;
      a[0] = arow[k + khalf];
      a[1] = arow[k + khalf + 1];
      bb[0] = brow[k + khalf];
      bb[1] = brow[k + khalf + 1];
      acc = __builtin_amdgcn_wmma_f32_16x16x4_f32(false, a, false, bb,
                                                  (short)0, acc, false, false);
    }
    __syncthreads();
  }

  // lanes 0-15 hold rows m = 0..7 (the 8 real heads) in acc[0..7]
  if (lane < 16) {
    float mx = acc[0];
#pragma unroll
    for (int v = 1; v < 8; ++v) mx = fmaxf(mx, acc[v]);
    l_out[(size_t)b * C_ + c0 + wave * 16 + lm] = mx;
  }
}

// ---------------------------------------------------------------------------
extern "C" void kernel_launch(void* const* d_in, const int* in_sizes, int n_in,
                              void* d_out, int out_size, void* d_ws, size_t ws_size,
                              hipStream_t stream) {
  const float* inputs      = (const float*)d_in[0];
  const float* W_att       = (const float*)d_in[1];
  const float* b_att       = (const float*)d_in[2];
  const float* W_clf       = (const float*)d_in[3];
  const float* b_clf       = (const float*)d_in[4];
  const float* W_enc       = (const float*)d_in[5];
  const float* b_enc       = (const float*)d_in[6];
  const float* embed_table = (const float*)d_in[7];
  const int*   group_y     = (const int*)d_in[8];

  // workspace: out [512x768] then emb [512x256]
  float* out_ws = (float*)d_ws;
  float* emb_ws = out_ws + (size_t)M_ * H_;

  // outputs, concatenated flat in return order
  float* c_out      = (float*)d_out;              // [64, 8192]
  float* l_out      = c_out + (size_t)B_ * GNUM_; // [64, 640]
  int*   candidates = (int*)(l_out + (size_t)B_ * C_);
  float* gscores    = (float*)(candidates + (size_t)B_ * C_);

  // 1) attention pooling -> out_ws
  attention_kernel<<<B_, 256, 0, stream>>>(inputs, W_att, b_att, out_ws);

  // 2) group classifier GEMM (512 x 8192 x 768) fused with max-over-heads
  {
    dim3 grid(GNUM_ / 64, M_ / 16);
    gemm_wmma_f32<1><<<grid, 128, 0, stream>>>(out_ws, W_clf, b_clf, c_out,
                                               M_, GNUM_, H_);
  }

  // 3) top-10 groups -> candidates + scores
  topk_kernel<<<B_, 256, 0, stream>>>(c_out, group_y, candidates, gscores);

  // 4) encoder GEMM (512 x 256 x 768) -> emb_ws
  {
    dim3 grid(E_ / 64, M_ / 16);
    gemm_wmma_f32<0><<<grid, 128, 0, stream>>>(out_ws, W_enc, b_enc, emb_ws,
                                               M_, E_, H_);
  }

  // 5) label scoring with gathered embeddings, max over heads -> l_out
  {
    dim3 grid(C_ / 64, B_);
    label_score_kernel<<<grid, 128, 0, stream>>>(emb_ws, embed_table,
                                                 candidates, l_out);
  }
}